// Sarf_19722489823700
// MI455X (gfx1250) — compile-verified
//
#include <hip/hip_runtime.h>

typedef __attribute__((ext_vector_type(2))) float v2f;
typedef __attribute__((ext_vector_type(8))) float v8f;

#define R_FRONT 80
#define R_ALL   160
#define H       1024
#define W       1024
#define NIMG    4
#define SEG_ROWS 64
#define NSEG    (H / SEG_ROWS)   /* 16 */

// -------------------------------------------------------------------------
// Pass 1: vertical sliding-window box sums (161 and 321 tall), zero padding.
// One thread per (image, column, 64-row segment); coalesced across columns.
// -------------------------------------------------------------------------
__global__ __launch_bounds__(256) void cfar_vert(const float* __restrict__ x,
                                                 float* __restrict__ c161,
                                                 float* __restrict__ c321) {
    int tid = blockIdx.x * blockDim.x + threadIdx.x;    // 0 .. 65535
    int col = tid & (W - 1);
    int seg = (tid >> 10) & (NSEG - 1);
    int img = tid >> 14;

    const float* base = x    + (size_t)img * H * W + col;
    float*       o161 = c161 + (size_t)img * H * W + col;
    float*       o321 = c321 + (size_t)img * H * W + col;

    const int y0 = seg * SEG_ROWS;

    // Halo init: windows centered at y0.
    float s161 = 0.f, s321 = 0.f;
    int dlo = y0 - R_ALL; if (dlo < 0) dlo = 0;
    int dhi = y0 + R_ALL; if (dhi > H - 1) dhi = H - 1;
    for (int d = dlo; d <= dhi; ++d) {
        float v = base[(size_t)d * W];
        s321 += v;
        int ad = d - y0;
        if (ad >= -R_FRONT && ad <= R_FRONT) s161 += v;
    }
    o161[(size_t)y0 * W] = s161;
    o321[(size_t)y0 * W] = s321;

    // Slide: O(1) per row.
    for (int y = y0 + 1; y < y0 + SEG_ROWS; ++y) {
        int ya = y + R_FRONT, yb = y - R_FRONT - 1;
        if (ya <= H - 1) s161 += base[(size_t)ya * W];
        if (yb >= 0)     s161 -= base[(size_t)yb * W];
        int yc = y + R_ALL, yd = y - R_ALL - 1;
        if (yc <= H - 1) s321 += base[(size_t)yc * W];
        if (yd >= 0)     s321 -= base[(size_t)yd * W];
        o161[(size_t)y * W] = s161;
        o321[(size_t)y * W] = s321;
    }
}

// -------------------------------------------------------------------------
// Pass 2: per-row prefix sums via V_WMMA_F32_16X16X4_F32 (P = L x V, L lower-
// triangular ones), carry fix-up, then window sums = prefix differences,
// then the CFAR ratio. One 128-thread block (4 waves) per row; each wave
// scans one 256-element chunk of both intermediate rows.
// -------------------------------------------------------------------------
__global__ __launch_bounds__(128) void cfar_horiz(const float* __restrict__ c161,
                                                  const float* __restrict__ c321,
                                                  float* __restrict__ out) {
    __shared__ float r161[W];
    __shared__ float r321[W];
    __shared__ float tot161[64], tot321[64];
    __shared__ float off161[64], off321[64];

    const int   row   = blockIdx.x;          // 0..4095 (img*1024 + y)
    const size_t rbase = (size_t)row * W;
    const int   t     = threadIdx.x;         // 0..127

    for (int i = t; i < W; i += 128) {
        r161[i] = c161[rbase + i];
        r321[i] = c321[rbase + i];
    }
    __syncthreads();

    const int lane = t & 31;
    const int wave = t >> 5;                 // chunk id 0..3
    const int half = lane >> 4;              // 0: lanes 0-15, 1: lanes 16-31
    const int n    = lane & 15;              // M for A-frag, N for B/C-frag

    const float* chA = &r161[wave * 256];
    const float* chB = &r321[wave * 256];

    // Chained WMMA: P = sum_kk  L[:,4kk:4kk+4] x V[4kk:4kk+4,:]
    // A 16x4 layout: lane<16 -> {K=0,K=1} in {v0,v1}; lane>=16 -> {K=2,K=3}.
    // B 4x16 layout: mirrored; C 16x16: VGPR r holds M = r + 8*half, N = lane&15.
    v8f p161 = {};
    v8f p321 = {};
#pragma unroll
    for (int kk = 0; kk < 4; ++kk) {
        const int kb = kk * 4 + half * 2;
        v2f a;  a[0] = (kb     <= n) ? 1.f : 0.f;
                a[1] = (kb + 1 <= n) ? 1.f : 0.f;
        v2f b1; b1[0] = chA[n * 16 + kb];  b1[1] = chA[n * 16 + kb + 1];
        v2f b2; b2[0] = chB[n * 16 + kb];  b2[1] = chB[n * 16 + kb + 1];
        p161 = __builtin_amdgcn_wmma_f32_16x16x4_f32(false, a, false, b1,
                                                     (short)0, p161, false, false);
        p321 = __builtin_amdgcn_wmma_f32_16x16x4_f32(false, a, false, b2,
                                                     (short)0, p321, false, false);
    }

    // Segment totals = row M=15 of P -> component 7 of lanes 16..31.
    if (half) {
        tot161[wave * 16 + n] = p161[7];
        tot321[wave * 16 + n] = p321[7];
    }
    __syncthreads();

    // Exclusive scan of the 64 segment totals (tiny, serial).
    if (t == 0) {
        float acc = 0.f;
        for (int s = 0; s < 64; ++s) { off161[s] = acc; acc += tot161[s]; }
    } else if (t == 1) {
        float acc = 0.f;
        for (int s = 0; s < 64; ++s) { off321[s] = acc; acc += tot321[s]; }
    }
    __syncthreads();

    // Scatter carry-corrected inclusive prefix back into LDS (in place).
    {
        const float o1 = off161[wave * 16 + n];
        const float o2 = off321[wave * 16 + n];
#pragma unroll
        for (int r = 0; r < 8; ++r) {
            const int mm = r + half * 8;                 // M index in segment
            r161[wave * 256 + n * 16 + mm] = p161[r] + o1;
            r321[wave * 256 + n * 16 + mm] = p321[r] + o2;
        }
    }
    __syncthreads();

    // Window sums from prefix differences + CFAR ratio (matches reference).
    const float BG_AREA_F   = 77120.f;            // 321^2 - 161^2
    const float FRONT_DIV_F = 25921.f * 1.8f;     // 161^2 * ALPHA
    for (int xx = t; xx < W; xx += 128) {
        int hi1 = xx + R_FRONT; if (hi1 > W - 1) hi1 = W - 1;
        int lo1 = xx - R_FRONT - 1;
        float front = r161[hi1] - (lo1 >= 0 ? r161[lo1] : 0.f);

        int hi2 = xx + R_ALL; if (hi2 > W - 1) hi2 = W - 1;
        int lo2 = xx - R_ALL - 1;
        float allsum = r321[hi2] - (lo2 >= 0 ? r321[lo2] : 0.f);

        float back = (allsum - front) / BG_AREA_F;
        float fr   = front / FRONT_DIV_F;
        out[rbase + xx] = fr / back;
    }
}

extern "C" void kernel_launch(void* const* d_in, const int* in_sizes, int n_in,
                              void* d_out, int out_size, void* d_ws, size_t ws_size,
                              hipStream_t stream) {
    const float* x   = (const float*)d_in[0];
    float*       out = (float*)d_out;
    float*       c161 = (float*)d_ws;                       // 16 MB
    float*       c321 = c161 + (size_t)NIMG * H * W;        // +16 MB

    const int vert_threads = NIMG * W * NSEG;               // 65536
    cfar_vert<<<vert_threads / 256, 256, 0, stream>>>(x, c161, c321);
    cfar_horiz<<<NIMG * H, 128, 0, stream>>>(c161, c321, out);
}